// Unpool_wSkip_9500467658972
// MI455X (gfx1250) — compile-verified
//
#include <hip/hip_runtime.h>
#include <hip/hip_bf16.h>

typedef __attribute__((ext_vector_type(2))) float v2f;
typedef __attribute__((ext_vector_type(8))) float v8f;

// ---------------------------------------------------------------------------
// Kernel 1: (optional gather+maxpool) -> X(rows,64) @ W(64,64) + bias,
// store Y, accumulate per-channel sum / sumsq for BatchNorm stats.
// Block = 256 threads = 8 waves; each wave computes a 16x64 output tile with
// V_WMMA_F32_16X16X4_F32 chained over K=64 (16 steps of K=4).
//
// fp32 WMMA operand layouts (ISA 7.12.2):
//   A 16x4:  lanes 0-15 (M=lane) hold K=kb+0 (v0), kb+1 (v1);
//            lanes 16-31 (M=lane-16) hold K=kb+2 (v0), kb+3 (v1)
//   B 4x16:  v0 = rows K=kb+0 (lanes 0-15) / kb+2 (lanes 16-31), v1 = +1
//   C/D 16x16: vgpr v, lanes 0-15 -> (M=v, N=lane); lanes 16-31 -> (M=v+8)
//
// W is staged into LDS pre-swizzled into B-fragment order: sWB[t][c][lane]
// is the float2 fragment for K-step t, column tile c, lane L.  Each inner-
// loop fragment fetch is then a single ds_load_b64 with an immediate offset,
// landing directly in an even-aligned VGPR pair (no re-pack moves).
// ---------------------------------------------------------------------------
template <bool GATHER>
__global__ __launch_bounds__(256)
void gemm_bn_stats(const float* __restrict__ X,    // GATHER: curr_feat (B,M,64) else skip_feat (rows,64)
                   const int*   __restrict__ idx,  // (rows,16) or nullptr
                   const float* __restrict__ W,    // (64,64) row-major (K,O)
                   const float* __restrict__ bias, // (64)
                   float*       __restrict__ Y,    // (rows,64)
                   float*       __restrict__ stats,// sum[64], sumsq[64]
                   int N, int M)                   // per-batch skip rows / curr rows
{
    __shared__ v2f   sWB[16 * 4 * 32];   // 16 KB: B fragments [t][c][lane]
    __shared__ float sB[64];
    __shared__ float red[128];

    const int tid = threadIdx.x;

    // ---- Stage W swizzled into B-fragment layout --------------------------
    for (int i = tid; i < 16 * 4 * 32; i += 256) {
        const int t  = i >> 7;           // k-step
        const int c  = (i >> 5) & 3;     // column tile
        const int ln = i & 31;           // lane
        const int k  = 4 * t + ((ln >> 4) << 1);
        const int n  = c * 16 + (ln & 15);
        v2f f;
        f.x = W[k * 64 + n];
        f.y = W[(k + 1) * 64 + n];
        sWB[i] = f;
    }
    if (tid < 64) sB[tid] = bias[tid];
    if (tid < 128) red[tid] = 0.0f;
    __syncthreads();

    const int wave  = tid >> 5;
    const int lane  = tid & 31;
    const int lcol  = lane & 15;         // column / A-row within tile
    const int half2 = (lane >> 4) << 1;  // 0 or 2 : K sub-offset for this half-wave
    const int rowbase = (blockIdx.x * 8 + wave) * 16;
    const int row     = rowbase + lcol;  // this lane's A-matrix row

    // ---- Build A fragments for all 16 K-steps (32 floats / lane) ----------
    v2f a[16];
    if (GATHER) {
        const int b = row / N;
        const float* fb = X + (size_t)b * M * 64;
        const int* ip = idx + (size_t)row * 16;
        #pragma unroll
        for (int t = 0; t < 16; ++t) a[t] = (v2f){-3.402823466e38f, -3.402823466e38f};
        for (int k = 0; k < 16; ++k) {
            const int m = ip[k];
            const float* src = fb + (size_t)m * 64 + half2;
            #pragma unroll
            for (int t = 0; t < 16; ++t) {
                v2f v = *(const v2f*)(src + 4 * t);
                a[t].x = fmaxf(a[t].x, v.x);
                a[t].y = fmaxf(a[t].y, v.y);
            }
        }
    } else {
        const float* src = X + (size_t)row * 64 + half2;
        #pragma unroll
        for (int t = 0; t < 16; ++t) a[t] = *(const v2f*)(src + 4 * t);
    }

    // ---- K-chained fp32 WMMA over 4 column tiles --------------------------
    const v2f* fragBase = sWB + lane;    // + (t*4+c)*32 as immediate DS offset
    v8f acc[4] = {};
    #pragma unroll
    for (int t = 0; t < 16; ++t) {
        #pragma unroll
        for (int c = 0; c < 4; ++c) {
            const v2f bf = fragBase[(t * 4 + c) * 32];   // one ds_load_b64
            acc[c] = __builtin_amdgcn_wmma_f32_16x16x4_f32(
                false, a[t], false, bf, (short)0, acc[c], false, false);
        }
    }

    // ---- bias, store Y, per-channel partial stats -------------------------
    const int mofs = (lane >> 4) << 3;   // 0 or 8 : D-matrix row offset
    #pragma unroll
    for (int c = 0; c < 4; ++c) {
        const float bv = sB[c * 16 + lcol];
        float ps = 0.0f, psq = 0.0f;
        #pragma unroll
        for (int v = 0; v < 8; ++v) {
            const float val = acc[c][v] + bv;
            Y[(size_t)(rowbase + mofs + v) * 64 + c * 16 + lcol] = val;
            ps  += val;
            psq += val * val;
        }
        atomicAdd(&red[c * 16 + lcol], ps);          // ds_add_f32
        atomicAdd(&red[64 + c * 16 + lcol], psq);
    }
    __syncthreads();
    if (tid < 128) unsafeAtomicAdd(&stats[tid], red[tid]);  // global_atomic_add_f32
}

// ---------------------------------------------------------------------------
__global__ void zero_stats(float* ws) { ws[threadIdx.x] = 0.0f; }

__global__ void finalize_stats(const float* __restrict__ stats,   // 2 x {sum[64],sumsq[64]}
                               const float* __restrict__ g1, const float* __restrict__ be1,
                               const float* __restrict__ g2, const float* __restrict__ be2,
                               float* __restrict__ derived,       // 2 x {scale[64],shift[64]}
                               float invBN)
{
    const int t = threadIdx.x;           // 128
    const int grp = t >> 6, c = t & 63;
    const float* s = stats + grp * 128;
    const float mean = s[c] * invBN;
    const float var  = s[64 + c] * invBN - mean * mean;
    const float gm = grp ? g2[c] : g1[c];
    const float bt = grp ? be2[c] : be1[c];
    const float sc = rsqrtf(var + 1e-5f) * gm;
    derived[grp * 128 + c]      = sc;
    derived[grp * 128 + 64 + c] = bt - mean * sc;
}

__global__ void writeout(const float* __restrict__ skip_coords,
                         const float* __restrict__ y1,
                         const float* __restrict__ y2,
                         const float* __restrict__ derived,
                         float* __restrict__ out, int total)
{
    const int i = blockIdx.x * blockDim.x + threadIdx.x;
    if (i >= total) return;
    const unsigned row = (unsigned)i / 131u;
    const unsigned j   = (unsigned)i - row * 131u;
    float v;
    if (j < 3u) {
        v = skip_coords[(size_t)row * 3 + j];
    } else if (j < 67u) {
        const unsigned c = j - 3u;
        v = fmaxf(y1[(size_t)row * 64 + c] * derived[c] + derived[64 + c], 0.0f);
    } else {
        const unsigned c = j - 67u;
        v = fmaxf(y2[(size_t)row * 64 + c] * derived[128 + c] + derived[192 + c], 0.0f);
    }
    out[i] = v;
}

// ---------------------------------------------------------------------------
extern "C" void kernel_launch(void* const* d_in, const int* in_sizes, int n_in,
                              void* d_out, int out_size, void* d_ws, size_t ws_size,
                              hipStream_t stream)
{
    const float* curr_feat   = (const float*)d_in[1];
    const float* skip_coords = (const float*)d_in[2];
    const float* skip_feat   = (const float*)d_in[3];
    const int*   up_idx      = (const int*)d_in[4];
    const float* W1  = (const float*)d_in[5];
    const float* b1  = (const float*)d_in[6];
    const float* g1  = (const float*)d_in[7];
    const float* be1 = (const float*)d_in[8];
    const float* W2  = (const float*)d_in[9];
    const float* b2  = (const float*)d_in[10];
    const float* g2  = (const float*)d_in[11];
    const float* be2 = (const float*)d_in[12];

    const int B = 4;
    const int N = (in_sizes[2] / 3) / B;   // 32768
    const int M = (in_sizes[0] / 3) / B;   // 16384
    const int rows = B * N;                // 131072

    float* ws      = (float*)d_ws;
    float* stats1  = ws;                 // 128 floats
    float* stats2  = ws + 128;           // 128 floats
    float* derived = ws + 256;           // 256 floats
    float* y1      = ws + 512;
    float* y2      = y1 + (size_t)rows * 64;

    zero_stats<<<1, 256, 0, stream>>>(ws);

    const int blocks = rows / 128;       // 8 waves x 16 rows per block
    gemm_bn_stats<true ><<<blocks, 256, 0, stream>>>(curr_feat, up_idx, W1, b1, y1, stats1, N, M);
    gemm_bn_stats<false><<<blocks, 256, 0, stream>>>(skip_feat, nullptr, W2, b2, y2, stats2, N, M);

    finalize_stats<<<1, 128, 0, stream>>>(ws, g1, be1, g2, be2, derived, 1.0f / (float)rows);

    const int total = rows * 131;
    writeout<<<(total + 255) / 256, 256, 0, stream>>>(skip_coords, y1, y2, derived,
                                                      (float*)d_out, total);
}